// ComplexExponentialPositionalEncoding_60997125538046
// MI455X (gfx1250) — compile-verified
//
#include <hip/hip_runtime.h>
#include <math.h>

typedef __attribute__((ext_vector_type(2))) float v2f;
typedef __attribute__((ext_vector_type(4))) float v4f;
typedef __attribute__((ext_vector_type(8))) float v8f;

// Reference constants (H, W fixed by the reference file).
constexpr int   H  = 384;
constexpr int   W  = 384;
constexpr int   D  = 1024;   // EMBED_DIM
constexpr int   DH = 512;    // D / 2 frequencies
constexpr float INV_MAXSEQ = 1.0f / 1000.0f;
constexpr float ALPHA      = 0.1f;
constexpr int   TJ = 8;      // j values per block in the streaming kernel

// ---------------------------------------------------------------------------
// Phase 1: build enc tables. ang[m,n] = pos[m] * freqs[n] is a rank-1 outer
// product -> compute each 16x16 tile with V_WMMA_F32_16X16X4_F32 (K=1, rest
// zero-padded). Per ISA layout: A(16x4) VGPR0 lanes0-15 = K=0 column;
// B(4x16) VGPR0 lanes0-15 = K=0 row. D(16x16): elem(M = v + 8*hi, N = lane&15)
// in accumulator VGPR v. One wave (32 threads) per tile; EXEC all ones.
// ---------------------------------------------------------------------------
__global__ void pe_encode_wmma(const float* __restrict__ freqs_i,
                               const float* __restrict__ freqs_j,
                               float* __restrict__ ws) {
  const int  lane  = threadIdx.x;   // 0..31, wave32
  const int  tn    = blockIdx.x;    // frequency tile, 0..DH/16-1
  const int  tm    = blockIdx.y;    // position tile,  0..H/16-1
  const int  which = blockIdx.z;    // 0 = enc_i, 1 = enc_j
  const float* __restrict__ freqs = which ? freqs_j : freqs_i;
  float* __restrict__ enc = ws + (size_t)which * H * D;

  const int  n  = lane & 15;
  const bool hi = lane >= 16;

  v2f a; a.x = hi ? 0.0f : (float)(tm * 16 + lane);  a.y = 0.0f;  // A: 16x4, K=0 col
  v2f b; b.x = hi ? 0.0f : freqs[tn * 16 + n];       b.y = 0.0f;  // B: 4x16, K=0 row
  v8f c = {};
  c = __builtin_amdgcn_wmma_f32_16x16x4_f32(
      /*neg_a=*/false, a, /*neg_b=*/false, b,
      /*c_mod=*/(short)0, c, /*reuse_a=*/false, /*reuse_b=*/false);

#pragma unroll
  for (int v = 0; v < 8; ++v) {
    const int   m      = v + (hi ? 8 : 0);
    const int   posIdx = tm * 16 + m;
    const float ang    = c[v];
    float s, co;
    sincosf(ang, &s, &co);
    v2f o; o.x = co; o.y = s;                 // interleaved [cos, sin]
    *(v2f*)(enc + (size_t)posIdx * D + 2 * (tn * 16 + n)) = o;
  }
}

// ---------------------------------------------------------------------------
// Phase 2: out[i,j,:] = (enc_i[i,:] + enc_j[j,:]) * scale(i,j).
// One block per (i, 8 consecutive j). Each thread owns a float4 of the
// enc_i row in registers; per j: L2-resident float4 load of enc_j, scalar
// scale, FMA, non-temporal B128 store (604 MB output won't be re-read).
// ---------------------------------------------------------------------------
__global__ void pe_bcast_scale(const float* __restrict__ enc_i,
                               const float* __restrict__ enc_j,
                               float* __restrict__ out) {
  const int i  = blockIdx.y;
  const int j0 = blockIdx.x * TJ;
  const int k  = threadIdx.x * 4;           // 256 threads cover D=1024 floats
  const float fi = (float)i * INV_MAXSEQ;

  const v4f ei = *(const v4f*)(enc_i + (size_t)i * D + k);

#pragma unroll
  for (int t = 0; t < TJ; ++t) {
    const int j  = j0 + t;
    const int jn = (j + 1 < W) ? (j + 1) : j;
    __builtin_prefetch(enc_j + (size_t)jn * D + k, 0, 0);  // global_prefetch_b8

    const float fj    = (float)j * INV_MAXSEQ;
    const float scale = 1.0f + ALPHA * sqrtf(fi * fi + fj * fj);

    const v4f ej = *(const v4f*)(enc_j + (size_t)j * D + k);
    const v4f o  = (ei + ej) * scale;
    __builtin_nontemporal_store(o, (v4f*)(out + ((size_t)i * W + j) * D + k));
  }
}

// ---------------------------------------------------------------------------
// Fallback (only if workspace is too small for the 3 MB enc tables):
// fully fused, recomputes sincos per element. Correct but slower.
// ---------------------------------------------------------------------------
__global__ void pe_fused(const float* __restrict__ freqs_i,
                         const float* __restrict__ freqs_j,
                         float* __restrict__ out) {
  const int i = blockIdx.y;
  const int j = blockIdx.x;
  const float fi = (float)i * INV_MAXSEQ;
  const float fj = (float)j * INV_MAXSEQ;
  const float scale = 1.0f + ALPHA * sqrtf(fi * fi + fj * fj);
  for (int p = threadIdx.x; p < DH; p += blockDim.x) {
    const float ai = (float)i * freqs_i[p];
    const float aj = (float)j * freqs_j[p];
    float si, ci, sj, cj;
    sincosf(ai, &si, &ci);
    sincosf(aj, &sj, &cj);
    v2f o; o.x = (ci + cj) * scale; o.y = (si + sj) * scale;
    __builtin_nontemporal_store(o, (v2f*)(out + ((size_t)i * W + j) * D + 2 * p));
  }
}

extern "C" void kernel_launch(void* const* d_in, const int* in_sizes, int n_in,
                              void* d_out, int out_size, void* d_ws, size_t ws_size,
                              hipStream_t stream) {
  // setup_inputs order: height(int,1), width(int,1), freqs_i(512 f32), freqs_j(512 f32)
  const float* freqs_i = (const float*)d_in[2];
  const float* freqs_j = (const float*)d_in[3];
  float* out = (float*)d_out;

  const size_t encBytes = 2ull * H * D * sizeof(float);  // ~3 MB
  if (ws_size >= encBytes) {
    float* ws = (float*)d_ws;
    // Phase 1: 32 x 24 x 2 tiles, one wave each.
    pe_encode_wmma<<<dim3(DH / 16, H / 16, 2), 32, 0, stream>>>(freqs_i, freqs_j, ws);
    // Phase 2: streaming broadcast + scale.
    pe_bcast_scale<<<dim3(W / TJ, H), 256, 0, stream>>>(ws, ws + (size_t)H * D, out);
  } else {
    pe_fused<<<dim3(W, H), 256, 0, stream>>>(freqs_i, freqs_j, out);
  }
}